// CrossHeadProjection_8272107012159
// MI455X (gfx1250) — compile-verified
//
#include <hip/hip_runtime.h>

typedef float v2f __attribute__((ext_vector_type(2)));
typedef float v4f __attribute__((ext_vector_type(4)));
typedef float v8f __attribute__((ext_vector_type(8)));

constexpr int T  = 2048;
constexpr int S  = 2048;
constexpr int M  = 32;                 // heads (G=1, M=H)
constexpr unsigned TS = 2048u * 2048u; // T*S in elements (byte offsets fit i32)
constexpr int SB = 128;                // s-positions per block (8 waves x 16 columns)
constexpr int TB = 16;                 // t-positions per block
constexpr int NT = 256;                // threads per block (8 waves)
constexpr int KW_STRIDE = 68;          // padded dword stride for [s][m*2+i]
constexpr int KD_STRIDE = 36;          // padded dword stride for [s][m]
constexpr int MT_STRIDE = 36;          // padded dword stride for Mt rows

// exchange with the other lane-half (lane ^ 16) via immediate-pattern ds_swizzle (SWAPX16)
__device__ __forceinline__ float swap16(float v) {
  return __builtin_bit_cast(float,
      __builtin_amdgcn_ds_swizzle(__builtin_bit_cast(int, v), 0x401F));
}

// 16 x-row loads in B-fragment layout; base already contains 2*hi*TS + t*S + s
__device__ __forceinline__ void load_b(const float* __restrict__ x, unsigned base,
                                       float b[8][2]) {
  #pragma unroll
  for (int k = 0; k < 8; ++k)
    #pragma unroll
    for (int j = 0; j < 2; ++j)
      b[k][j] = __builtin_nontemporal_load(x + (base + (unsigned)(4 * k + j) * TS));
}

// 16 WMMAs: two 16-row output halves x 8 K-steps (K = 32 heads)
__device__ __forceinline__ void mma32(const v2f aLo[8], const v2f aHi[8],
                                      const float b[8][2], v8f& d0, v8f& d1) {
  #pragma unroll
  for (int k = 0; k < 8; ++k) {
    v2f bb; bb.x = b[k][0]; bb.y = b[k][1];
    d0 = __builtin_amdgcn_wmma_f32_16x16x4_f32(false, aLo[k], false, bb, (short)0, d0, false, false);
    d1 = __builtin_amdgcn_wmma_f32_16x16x4_f32(false, aHi[k], false, bb, (short)0, d1, false, false);
  }
}

// per-s rank-2 (kw) + diagonal (kdd) tail, then NT store; obase contains 8*hi*TS + t*S + s
__device__ __forceinline__ void tail_store(float* __restrict__ out, unsigned obase,
                                           const float b[8][2], const v8f& d0, const v8f& d1,
                                           const float* __restrict__ kw1row,
                                           const float* __restrict__ kw2row,
                                           const float* __restrict__ kddrow, int hi) {
  float part0 = 0.f, part1 = 0.f;
  #pragma unroll
  for (int k = 0; k < 8; ++k)
    #pragma unroll
    for (int j = 0; j < 2; ++j) {
      v2f kv = *(const v2f*)&kw1row[(4 * k + 2 * hi + j) * 2];
      part0 += kv.x * b[k][j];
      part1 += kv.y * b[k][j];
    }
  float hk0 = part0 + swap16(part0);      // other lane-half holds the other 16 heads
  float hk1 = part1 + swap16(part1);

  float bx[8][2];                         // other-half x values for the kdd diagonal
  #pragma unroll
  for (int k = 0; k < 8; ++k)
    #pragma unroll
    for (int j = 0; j < 2; ++j)
      bx[k][j] = swap16(b[k][j]);         // = x[4k+2*(1-hi)+j][s]

  #pragma unroll
  for (int h = 0; h < 2; ++h)
    #pragma unroll
    for (int r = 0; r < 8; ++r) {
      const int q = h * 16 + 8 * hi + r;  // output head row (D layout)
      v2f  w2 = *(const v2f*)&kw2row[q * 2];
      float kd = kddrow[q];
      const int r4 = r >> 2, rj = r & 1;  // q>>2 = 4h + 2hi + r4 ; owner half = (r>>1)&1
      float xq;
      if (((r >> 1) & 1) == 0)
        xq = hi ? bx[h * 4 + 2 + r4][rj] : b[h * 4 + r4][rj];
      else
        xq = hi ? b[h * 4 + 2 + r4][rj] : bx[h * 4 + r4][rj];
      float val = (h == 0 ? d0[r] : d1[r]) + w2.x * hk0 + w2.y * hk1 + kd * xq;
      __builtin_nontemporal_store(val, out + (obase + (unsigned)(h * 16 + r) * TS));
    }
}

__global__ __launch_bounds__(NT)
void chp_kernel(const float* __restrict__ x,   const float* __restrict__ w,
                const float* __restrict__ qw1, const float* __restrict__ qw2,
                const float* __restrict__ kw1, const float* __restrict__ kw2,
                const float* __restrict__ qdd, const float* __restrict__ kdd,
                float* __restrict__ out) {
  __shared__ float smKw1[SB * KW_STRIDE];
  __shared__ float smKw2[SB * KW_STRIDE];
  __shared__ float smKdd[SB * KD_STRIDE];
  __shared__ float smMt [M * MT_STRIDE];
  __shared__ float smW  [M * M];
  __shared__ float smQw1[TB * M * 2];
  __shared__ float smQw2[TB * M * 2];
  __shared__ float smQdd[TB * M];

  const int tid = threadIdx.x;
  const int sB  = blockIdx.x * SB;
  const int tB  = blockIdx.y * TB;

  // ---- one-time staging: per-s weights (padded rows), per-t weights + w (flat) ----
  {
    const v4f* g1 = (const v4f*)(kw1 + (size_t)sB * (M * 2));
    const v4f* g2 = (const v4f*)(kw2 + (size_t)sB * (M * 2));
    for (int i = tid; i < SB * M * 2 / 4; i += NT) {
      int s = i >> 4, c = (i & 15) * 4;            // 16 float4 per 64-float s-row
      *(v4f*)&smKw1[s * KW_STRIDE + c] = g1[i];
      *(v4f*)&smKw2[s * KW_STRIDE + c] = g2[i];
    }
    const v4f* g3 = (const v4f*)(kdd + (size_t)sB * M);
    for (int i = tid; i < SB * M / 4; i += NT) {
      int s = i >> 3, c = (i & 7) * 4;             // 8 float4 per 32-float s-row
      *(v4f*)&smKdd[s * KD_STRIDE + c] = g3[i];
    }
    const v4f* gw = (const v4f*)w;
    for (int i = tid; i < M * M / 4; i += NT)
      *(v4f*)&smW[i * 4] = gw[i];
    const v4f* q1 = (const v4f*)(qw1 + (size_t)tB * (M * 2));
    const v4f* q2 = (const v4f*)(qw2 + (size_t)tB * (M * 2));
    for (int i = tid; i < TB * M * 2 / 4; i += NT) {
      *(v4f*)&smQw1[i * 4] = q1[i];
      *(v4f*)&smQw2[i * 4] = q2[i];
    }
    const v4f* qd = (const v4f*)(qdd + (size_t)tB * M);
    for (int i = tid; i < TB * M / 4; i += NT)
      *(v4f*)&smQdd[i * 4] = qd[i];
  }

  const int lane = tid & 31;
  const int wv   = tid >> 5;           // 8 waves
  const int p    = lane & 15;          // column slot within the wave's 16-wide group
  const int hi   = lane >> 4;          // lane half: selects K/row sub-block
  const int s0   = sB + wv * 16;       // each wave owns 16 contiguous s-columns
  const int sl0  = wv * 16 + p;        // local s index in the LDS tiles

  // 32-bit element offsets (uniform SGPR base + VGPR32 offset addressing)
  unsigned xb = (unsigned)(2 * hi) * TS + (unsigned)(tB * S) + (unsigned)(s0 + p);
  unsigned ob = (unsigned)(8 * hi) * TS + (unsigned)(tB * S) + (unsigned)(s0 + p);

  const float* kw1r = &smKw1[sl0 * KW_STRIDE];
  const float* kw2r = &smKw2[sl0 * KW_STRIDE];
  const float* kddr = &smKdd[sl0 * KD_STRIDE];

  for (int tt = 0; tt < TB; ++tt) {
    __syncthreads();   // previous iteration's readers of smMt are done

    // ---- Mt(t)[n][m] = delta(n,m)*(1+qdd) + w[m][n] + sum_i qw2[n,i]*qw1[m,i] (all LDS) ----
    for (int idx = tid; idx < M * M; idx += NT) {
      int n = idx >> 5, m = idx & 31;
      float v = smW[m * M + n]
              + smQw2[tt * 64 + n * 2 + 0] * smQw1[tt * 64 + m * 2 + 0]
              + smQw2[tt * 64 + n * 2 + 1] * smQw1[tt * 64 + m * 2 + 1];
      if (n == m) v += 1.0f + smQdd[tt * 32 + n];
      smMt[n * MT_STRIDE + m] = v;
    }
    __syncthreads();

    // ---- A fragments of Mt ----
    v2f aLo[8], aHi[8];
    #pragma unroll
    for (int k = 0; k < 8; ++k) {
      aLo[k] = *(const v2f*)&smMt[p        * MT_STRIDE + 4 * k + 2 * hi];
      aHi[k] = *(const v2f*)&smMt[(16 + p) * MT_STRIDE + 4 * k + 2 * hi];
    }

    // ---- load x fragments, 16 WMMAs, tail, store ----
    float b0[8][2];
    load_b(x, xb, b0);

    v8f d0 = {}, d1 = {};
    mma32(aLo, aHi, b0, d0, d1);
    tail_store(out, ob, b0, d0, d1, kw1r, kw2r, kddr, hi);

    xb += (unsigned)S;   // next t
    ob += (unsigned)S;
  }
}

extern "C" void kernel_launch(void* const* d_in, const int* in_sizes, int n_in,
                              void* d_out, int out_size, void* d_ws, size_t ws_size,
                              hipStream_t stream) {
  const float* x   = (const float*)d_in[0];
  const float* w   = (const float*)d_in[1];
  const float* qw1 = (const float*)d_in[2];
  const float* qw2 = (const float*)d_in[3];
  const float* kw1 = (const float*)d_in[4];
  const float* kw2 = (const float*)d_in[5];
  const float* qdd = (const float*)d_in[6];
  const float* kdd = (const float*)d_in[7];
  dim3 grid(S / SB, T / TB);   // 16 x 128 = 2048 blocks
  chp_kernel<<<grid, dim3(NT), 0, stream>>>(x, w, qw1, qw2, kw1, kw2, qdd, kdd,
                                            (float*)d_out);
}